// MyGaussianEnsembleFast_1623497637994
// MI455X (gfx1250) — compile-verified
//
#include <hip/hip_runtime.h>
#include <hip/hip_bf16.h>
#include <math.h>

typedef __attribute__((ext_vector_type(2))) float v2f;
typedef __attribute__((ext_vector_type(8))) float v8f;
typedef __attribute__((ext_vector_type(4))) unsigned int v4u;
typedef __attribute__((ext_vector_type(8))) int v8i;
typedef __attribute__((ext_vector_type(4))) int v4i;

#define NPIX 4096
#define CHUNK 64
#define NROWS 12   // rows 0..5 Psi, 6..7 zero pad, 8..10 rgb, 11 zero

__device__ __forceinline__ float sigmoidf_(float x) { return 1.0f / (1.0f + __expf(-x)); }

// ---------------------------------------------------------------------------
// Kernel 1: conv3x3(3->64) + head(64->36) + per-gaussian mixture reduction.
// Emits one 12x4096 coefficient tensor per batch image (SoA over n):
//   rows 0..5 : Psi (quadratic-form coefficients of sigma(p,n) = Phi(p).Psi(n))
//   rows 6..7 : zero (K padding for WMMA K=8)
//   rows 8..10: weighted rgb, row 11: zero (channel padding)
// ---------------------------------------------------------------------------
__global__ void __launch_bounds__(256) param_kernel(
    const float* __restrict__ inp, const float* __restrict__ enc_w,
    const float* __restrict__ enc_b, const float* __restrict__ head_w,
    const float* __restrict__ head_b, float* __restrict__ coef_g) {
  __shared__ float sW[1728];   // enc_w  64*3*3*3
  __shared__ float sB[64];
  __shared__ float sHW[2304];  // head_w 36*64
  __shared__ float sHB[64];    // head_b (36 used)

  int tid = threadIdx.x;
  for (int i = tid; i < 1728; i += 256) sW[i] = enc_w[i];
  for (int i = tid; i < 2304; i += 256) sHW[i] = head_w[i];
  if (tid < 64) sB[tid] = enc_b[tid];
  if (tid < 64) sHB[tid] = (tid < 36) ? head_b[tid] : 0.0f;
  __syncthreads();

  int gid = blockIdx.x * 256 + tid;  // 0..8191
  int b = gid >> 12;
  int n = gid & (NPIX - 1);
  int h = n >> 6, w = n & 63;

  // 3x3 neighborhood, 3 input channels, zero padded
  float pix[27];
  const float* ib = inp + b * 3 * NPIX;
#pragma unroll
  for (int ic = 0; ic < 3; ++ic)
#pragma unroll
    for (int ky = 0; ky < 3; ++ky)
#pragma unroll
      for (int kx = 0; kx < 3; ++kx) {
        int hy = h + ky - 1, wx = w + kx - 1;
        bool ok = (hy >= 0) & (hy < 64) & (wx >= 0) & (wx < 64);
        pix[(ic * 3 + ky) * 3 + kx] = ok ? ib[ic * NPIX + hy * 64 + wx] : 0.0f;
      }

  float feat[64];
  for (int oc = 0; oc < 64; ++oc) {
    float acc = sB[oc];
    const float* wp = &sW[oc * 27];
#pragma unroll
    for (int q = 0; q < 27; ++q) acc = fmaf(wp[q], pix[q], acc);
    feat[oc] = acc;
  }

  float pred[36];
  for (int o = 0; o < 36; ++o) {
    float acc = sHB[o];
    const float* hp = &sHW[o * 64];
#pragma unroll 16
    for (int c = 0; c < 64; ++c) acc = fmaf(hp[c], feat[c], acc);
    pred[o] = acc;
  }

  // softmax over k of w_logits (pred[k*9+8])
  float wl0 = pred[8], wl1 = pred[17], wl2 = pred[26], wl3 = pred[35];
  float mx = fmaxf(fmaxf(wl0, wl1), fmaxf(wl2, wl3));
  float e0 = __expf(wl0 - mx), e1 = __expf(wl1 - mx);
  float e2 = __expf(wl2 - mx), e3 = __expf(wl3 - mx);
  float inv = 1.0f / (e0 + e1 + e2 + e3);
  float wk[4] = {e0 * inv, e1 * inv, e2 * inv, e3 * inv};

  float rgb0 = 0.f, rgb1 = 0.f, rgb2 = 0.f;
  float theta = 0.f, sx = 0.f, sy = 0.f, o0 = 0.f, o1 = 0.f;
#pragma unroll
  for (int k = 0; k < 4; ++k) {
    const float* pk = &pred[k * 9];
    float wkk = wk[k];
    rgb0 = fmaf(wkk, pk[0], rgb0);
    rgb1 = fmaf(wkk, pk[1], rgb1);
    rgb2 = fmaf(wkk, pk[2], rgb2);
    theta = fmaf(wkk, sigmoidf_(pk[3]) * 6.2831853071795864f, theta);
    sx = fmaf(wkk, sigmoidf_(pk[4]) * 0.5f + 1e-6f, sx);
    sy = fmaf(wkk, sigmoidf_(pk[5]) * 0.5f + 1e-6f, sy);
    o0 = fmaf(wkk, tanhf(pk[6]), o0);
    o1 = fmaf(wkk, tanhf(pk[7]), o1);
  }

  float sth, cth;
  __sincosf(theta, &sth, &cth);
  float isx = 1.0f / (sx * sx), isy = 1.0f / (sy * sy);
  float ca = cth * cth * isx + sth * sth * isy;
  float cb = cth * sth * (isx - isy);
  float cc = sth * sth * isx + cth * cth * isy;

  float u = 2.0f * (float)(n & 63) * (1.0f / 64.0f) - 1.0f;
  float v = 2.0f * (float)(n >> 6) * (1.0f / 64.0f) - 1.0f;
  float m0 = u + 2.0f * o0 * (1.0f / 64.0f) - (1.0f / 64.0f);
  float m1 = v + 2.0f * o1 * (1.0f / 64.0f) - (1.0f / 64.0f);

  float* pg = coef_g + b * NROWS * NPIX + n;
  pg[0 * NPIX] = 0.5f * ca;
  pg[1 * NPIX] = 0.5f * cc;
  pg[2 * NPIX] = cb;
  pg[3 * NPIX] = -(ca * m0 + cb * m1);
  pg[4 * NPIX] = -(cc * m1 + cb * m0);
  pg[5 * NPIX] = 0.5f * ca * m0 * m0 + 0.5f * cc * m1 * m1 + cb * m0 * m1;
  pg[6 * NPIX] = 0.0f;
  pg[7 * NPIX] = 0.0f;
  pg[8 * NPIX] = rgb0;
  pg[9 * NPIX] = rgb1;
  pg[10 * NPIX] = rgb2;
  pg[11 * NPIX] = 0.0f;
}

// ---------------------------------------------------------------------------
// Kernel 2: per-wave 16-pixel tile. sigma = Phi x Psi via V_WMMA_F32_16X16X4
// (K=8 -> 2 wmma), alpha = exp(-sigma) masked, img += alpha x rgb (4 wmma per
// 16-n subchunk, alpha relaid out through per-wave LDS tile).
// Coefficient chunks (12x64 tile of the 12x4096 tensor) are double-buffered
// into LDS by the Tensor Data Mover (tensor_load_to_lds + s_wait_tensorcnt).
// ---------------------------------------------------------------------------
__global__ void __launch_bounds__(256) render_kernel(
    const float* __restrict__ coef_g, float* __restrict__ out) {
  __shared__ float sC[2][NROWS][CHUNK];   // TDM-filled coefficient tiles
  __shared__ float sAlpha[8][16][17];     // [wave][p row][n col] padded

  int tid = threadIdx.x;
  int wave = tid >> 5;
  int lane = tid & 31;
  int m = lane & 15;
  int hi = lane >> 4;

  int b = blockIdx.x >> 5;                       // 32 blocks per batch image
  int pbase = (blockIdx.x & 31) * 128 + wave * 16;
  int p = pbase + m;

  float u = 2.0f * (float)(p & 63) * (1.0f / 64.0f) - 1.0f;
  float v = 2.0f * (float)(p >> 6) * (1.0f / 64.0f) - 1.0f;
  float phi[8] = {u * u, v * v, u * v, u, v, 1.0f, 0.0f, 0.0f};
  // A-matrix 16x4 f32 layout: lane m = row M, VGPR0 holds K=2*hi, VGPR1 K=2*hi+1
  v2f a0 = {phi[2 * hi], phi[2 * hi + 1]};
  v2f a1 = {phi[4 + 2 * hi], phi[5 + 2 * hi]};

  v8f cimg = {0.f, 0.f, 0.f, 0.f, 0.f, 0.f, 0.f, 0.f};

  const float* cB = coef_g + b * NROWS * NPIX;

  // TDM descriptor: 12x64 tile of the row-major 12x4096 coef tensor -> LDS.
  auto issue_tdm = [&](int nbase, int buf) {
    unsigned long long gaddr = (unsigned long long)(cB + nbase);
    unsigned ldsaddr = (unsigned)(unsigned long long)(&sC[buf][0][0]);
    v4u g0;
    g0.x = 1u;                                            // count=1 (valid)
    g0.y = ldsaddr;                                       // lds_addr
    g0.z = (unsigned)(gaddr & 0xFFFFFFFFull);             // global_addr lo
    g0.w = (unsigned)((gaddr >> 32) & 0x01FFFFFFull)      // global_addr hi
           | 0x80000000u;                                 // type=2 (image)
    v8i g1;
    g1[0] = 0x00020000;           // data_size=2 -> 4 bytes, no multicast/pad
    g1[1] = (int)(NPIX << 16);    // tensor_dim0[15:0]=4096 in bits 63:48
    g1[2] = (NROWS << 16);        // tensor_dim0 hi=0 | tensor_dim1[15:0]=12
    g1[3] = (CHUNK << 16);        // tensor_dim1 hi=0 | tile_dim0=64
    g1[4] = NROWS;                // tile_dim1=12, tile_dim2=0
    g1[5] = NPIX;                 // tensor_dim0_stride lo32 = 4096
    g1[6] = 0;                    // stride hi, tensor_dim1_stride lo
    g1[7] = 0;                    // tensor_dim1_stride hi (2D tensor)
    v4i zz4 = {0, 0, 0, 0};
    v8i zz8 = {0, 0, 0, 0, 0, 0, 0, 0};
    __builtin_amdgcn_tensor_load_to_lds(g0, g1, zz4, zz4, zz8, 0);
  };

  const int NCHUNKS = NPIX / CHUNK;
  if (tid < 32) issue_tdm(0, 0);          // prefetch chunk 0

  for (int c = 0; c < NCHUNKS; ++c) {
    int buf = c & 1;
    if (tid < 32) {
      if (c + 1 < NCHUNKS) {
        issue_tdm((c + 1) * CHUNK, buf ^ 1);   // prefetch next chunk
        __builtin_amdgcn_s_wait_tensorcnt((short)1);  // chunk c landed
      } else {
        __builtin_amdgcn_s_wait_tensorcnt((short)0);
      }
    }
    __syncthreads();   // chunk c visible to all waves

#pragma unroll
    for (int sub = 0; sub < CHUNK / 16; ++sub) {
      int nl = sub * 16;
      // B-matrix 4x16 f32 layout: VGPR0 row K=2*hi, VGPR1 row K=2*hi+1, col N=m
      v2f b0 = {sC[buf][2 * hi][nl + m], sC[buf][2 * hi + 1][nl + m]};
      v2f b1 = {sC[buf][4 + 2 * hi][nl + m], sC[buf][5 + 2 * hi][nl + m]};
      v8f sg = {0.f, 0.f, 0.f, 0.f, 0.f, 0.f, 0.f, 0.f};
      sg = __builtin_amdgcn_wmma_f32_16x16x4_f32(false, a0, false, b0,
                                                 (short)0, sg, false, false);
      sg = __builtin_amdgcn_wmma_f32_16x16x4_f32(false, a1, false, b1,
                                                 (short)0, sg, false, false);
      // alpha = sigma>=0 ? exp(-sigma) : 0 ; C/D layout -> LDS (row p, col n)
#pragma unroll
      for (int j = 0; j < 8; ++j) {
        float s = sg[j];
        sAlpha[wave][j + 8 * hi][m] = (s >= 0.0f) ? __expf(-s) : 0.0f;
      }
      // img(16p x 16ch) += alpha(16p x 16n) * rgb(16n x 16ch), K=16 -> 4 wmma
      int cIdx = (m < 4) ? (8 + m) : 11;   // row 11 of coef tile is zero
#pragma unroll
      for (int t = 0; t < 4; ++t) {
        v2f aa = {sAlpha[wave][m][4 * t + 2 * hi],
                  sAlpha[wave][m][4 * t + 2 * hi + 1]};
        v2f bb = {sC[buf][cIdx][nl + 4 * t + 2 * hi],
                  sC[buf][cIdx][nl + 4 * t + 2 * hi + 1]};
        cimg = __builtin_amdgcn_wmma_f32_16x16x4_f32(false, aa, false, bb,
                                                     (short)0, cimg, false, false);
      }
    }
    __syncthreads();   // everyone done with buf before TDM refills it
  }

  // C/D layout: lane -> channel m, VGPR j -> pixel pbase + j + 8*hi
  if (m < 3) {
    float* ob = out + (b * 3 + m) * NPIX;
#pragma unroll
    for (int j = 0; j < 8; ++j) {
      float val = cimg[j];
      val = fminf(fmaxf(val, 0.0f), 1.0f);
      ob[pbase + j + 8 * hi] = val;
    }
  }
}

extern "C" void kernel_launch(void* const* d_in, const int* in_sizes, int n_in,
                              void* d_out, int out_size, void* d_ws, size_t ws_size,
                              hipStream_t stream) {
  (void)in_sizes; (void)n_in; (void)out_size; (void)ws_size;
  const float* inp    = (const float*)d_in[0];
  const float* enc_w  = (const float*)d_in[1];
  const float* enc_b  = (const float*)d_in[2];
  const float* head_w = (const float*)d_in[3];
  const float* head_b = (const float*)d_in[4];

  float* coef_g = (float*)d_ws;   // 2 * 12 * 4096 floats = 384 KB
  float* outp   = (float*)d_out;  // 2 * 3 * 4096 floats

  param_kernel<<<32, 256, 0, stream>>>(inp, enc_w, enc_b, head_w, head_b, coef_g);
  render_kernel<<<64, 256, 0, stream>>>(coef_g, outp);
}